// GCNClassifier_8280696946778
// MI455X (gfx1250) — compile-verified
//
#include <hip/hip_runtime.h>
#include <math.h>

typedef float v2f __attribute__((ext_vector_type(2)));
typedef float v8f __attribute__((ext_vector_type(8)));

#define F_IN  128
#define F_OUT 64

// ---------------- degree / normalization ----------------

__global__ void k_init_deg(float* __restrict__ deg, int N) {
    int i = blockIdx.x * blockDim.x + threadIdx.x;
    if (i < N) deg[i] = 1.0f;  // self loop
}

__global__ void k_count_deg(const int* __restrict__ dst, float* __restrict__ deg, int E) {
    int e = blockIdx.x * blockDim.x + threadIdx.x;
    if (e < E) atomicAdd(&deg[dst[e]], 1.0f);
}

__global__ void k_dinv(float* __restrict__ deg, int N) {
    int i = blockIdx.x * blockDim.x + threadIdx.x;
    if (i < N) {
        float d = deg[i];
        deg[i] = (d > 0.0f) ? rsqrtf(d) : 0.0f;  // in place: deg -> dinv
    }
}

// ---------------- h = x @ W_conv^T via V_WMMA_F32_16X16X4_F32 ----------------
// One wave computes a 16x64 tile of h. A: 16x4 f32 (2 VGPR), B: 4x16 f32 (2 VGPR),
// C/D: 16x16 f32 (8 VGPR). Lanes 0-15 hold K={k0,k0+1}; lanes 16-31 hold K={k0+2,k0+3}.

__global__ void __launch_bounds__(256) k_gemm(const float* __restrict__ x,
                                              const float* __restrict__ W,
                                              float* __restrict__ h, int N) {
    const int wave = threadIdx.x >> 5;
    const int lane = threadIdx.x & 31;
    const int row0 = (blockIdx.x * 8 + wave) * 16;
    if (row0 >= N) return;

    const int lr    = lane & 15;       // M (for A) / N (for B) within tile
    const int khalf = lane >> 4;       // 0: K pair {0,1}; 1: K pair {2,3}

    int m = row0 + lr;
    if (m >= N) m = N - 1;             // clamp loads; stores guarded below

    const float2* xp  = (const float2*)(x + (size_t)m * F_IN);
    const float2* wp0 = (const float2*)(W + (size_t)(lr +  0) * F_IN);
    const float2* wp1 = (const float2*)(W + (size_t)(lr + 16) * F_IN);
    const float2* wp2 = (const float2*)(W + (size_t)(lr + 32) * F_IN);
    const float2* wp3 = (const float2*)(W + (size_t)(lr + 48) * F_IN);

    v8f c0 = {}, c1 = {}, c2 = {}, c3 = {};

    #pragma unroll
    for (int k0 = 0; k0 < F_IN; k0 += 4) {
        const int ki = (k0 >> 1) + khalf;   // float2 index covering K pair
        float2 af = xp[ki];
        v2f a; a.x = af.x; a.y = af.y;
        float2 b0f = wp0[ki]; v2f b0; b0.x = b0f.x; b0.y = b0f.y;
        float2 b1f = wp1[ki]; v2f b1; b1.x = b1f.x; b1.y = b1f.y;
        float2 b2f = wp2[ki]; v2f b2; b2.x = b2f.x; b2.y = b2f.y;
        float2 b3f = wp3[ki]; v2f b3; b3.x = b3f.x; b3.y = b3f.y;

        c0 = __builtin_amdgcn_wmma_f32_16x16x4_f32(false, a, false, b0, (short)0, c0, false, false);
        c1 = __builtin_amdgcn_wmma_f32_16x16x4_f32(false, a, false, b1, (short)0, c1, false, false);
        c2 = __builtin_amdgcn_wmma_f32_16x16x4_f32(false, a, false, b2, (short)0, c2, false, false);
        c3 = __builtin_amdgcn_wmma_f32_16x16x4_f32(false, a, false, b3, (short)0, c3, false, false);
    }

    // C/D layout: VGPR v, lanes 0-15 -> M=v, lanes 16-31 -> M=v+8; N = lane&15.
    const int mbase = row0 + (khalf << 3);
    #pragma unroll
    for (int v = 0; v < 8; ++v) {
        int mm = mbase + v;
        if (mm < N) {
            float* hp = h + (size_t)mm * F_OUT + lr;
            hp[ 0] = c0[v];
            hp[16] = c1[v];
            hp[32] = c2[v];
            hp[48] = c3[v];
        }
    }
}

// ---------------- agg init: self-loop contribution + bias ----------------

__global__ void k_init_agg(const float* __restrict__ h, const float* __restrict__ dinv,
                           const float* __restrict__ b_conv, float* __restrict__ agg, int total) {
    int idx = blockIdx.x * blockDim.x + threadIdx.x;
    if (idx < total) {
        int node = idx >> 6;
        int f    = idx & 63;
        float di = dinv[node];
        agg[idx] = b_conv[f] + h[idx] * di * di;
    }
}

// ---------------- edge scatter: one wave per edge, float2 per lane ----------------

__global__ void k_scatter(const int* __restrict__ src, const int* __restrict__ dst,
                          const float* __restrict__ h, const float* __restrict__ dinv,
                          float* __restrict__ agg, int E) {
    int gtid = blockIdx.x * blockDim.x + threadIdx.x;
    int e    = gtid >> 5;
    int lane = threadIdx.x & 31;
    if (e >= E) return;

    int s = src[e];
    int d = dst[e];
    float norm = dinv[s] * dinv[d];

    const float2* hs = (const float2*)(h + (size_t)s * F_OUT);
    float2 v = hs[lane];

    float* ad = agg + (size_t)d * F_OUT + lane * 2;
    atomicAdd(ad + 0, v.x * norm);
    atomicAdd(ad + 1, v.y * norm);
}

// ---------------- finalize: relu -> dot(W_lin) -> sigmoid ----------------

__global__ void k_finalize(const float* __restrict__ agg, const float* __restrict__ W_lin,
                           const float* __restrict__ b_lin, float* __restrict__ out, int N) {
    int gtid = blockIdx.x * blockDim.x + threadIdx.x;
    int node = gtid >> 5;
    int lane = threadIdx.x & 31;
    if (node >= N) return;

    const float* row = agg + (size_t)node * F_OUT;
    float p = fmaxf(row[lane], 0.0f) * W_lin[lane]
            + fmaxf(row[lane + 32], 0.0f) * W_lin[lane + 32];

    #pragma unroll
    for (int off = 16; off > 0; off >>= 1)
        p += __shfl_down(p, off);

    if (lane == 0) {
        float z = p + b_lin[0];
        out[node] = 1.0f / (1.0f + __expf(-z));
    }
}

// ---------------- launcher ----------------

extern "C" void kernel_launch(void* const* d_in, const int* in_sizes, int n_in,
                              void* d_out, int out_size, void* d_ws, size_t ws_size,
                              hipStream_t stream) {
    const float* x      = (const float*)d_in[0];
    const int*   edge   = (const int*)  d_in[1];
    const float* W_conv = (const float*)d_in[2];
    const float* b_conv = (const float*)d_in[3];
    const float* W_lin  = (const float*)d_in[4];
    const float* b_lin  = (const float*)d_in[5];
    float*       out    = (float*)d_out;

    const int N = in_sizes[0] / F_IN;
    const int E = in_sizes[1] / 2;
    const int* src = edge;
    const int* dst = edge + E;

    // Workspace layout (floats): [deg/dinv: N][h: N*64][agg: N*64]  ~= 52 MB
    float* ws   = (float*)d_ws;
    float* dinv = ws;
    float* h    = ws + N;
    float* agg  = h + (size_t)N * F_OUT;

    k_init_deg <<<(N + 255) / 256, 256, 0, stream>>>(dinv, N);
    k_count_deg<<<(E + 255) / 256, 256, 0, stream>>>(dst, dinv, E);
    k_dinv     <<<(N + 255) / 256, 256, 0, stream>>>(dinv, N);

    k_gemm<<<(N + 127) / 128, 256, 0, stream>>>(x, W_conv, h, N);

    const int tot = N * F_OUT;
    k_init_agg<<<(tot + 255) / 256, 256, 0, stream>>>(h, dinv, b_conv, agg, tot);

    const long long sth = (long long)E * 32;
    k_scatter<<<(int)((sth + 255) / 256), 256, 0, stream>>>(src, dst, h, dinv, agg, E);

    const long long fth = (long long)N * 32;
    k_finalize<<<(int)((fth + 255) / 256), 256, 0, stream>>>(agg, W_lin, b_lin, out, N);
}